// Conv3d_82669530514090
// MI455X (gfx1250) — compile-verified
//
#include <hip/hip_runtime.h>
#include <hip/hip_bf16.h>
#include <stdint.h>

// ---------------------------------------------------------------------------
// Sparse submanifold 3D conv for MI455X (gfx1250, wave32).
//   per offset k: gather x rows -> [16x32] x [32x32] fp32 GEMM via
//   V_WMMA_F32_16X16X4_F32 -> scatter-add with global_atomic_add_f32.
// x (32MB) and out (32MB) are L2-resident (192MB L2), so the random
// gather/scatter is L2-bound; fp32 WMMA keeps exact reference precision.
// Kernel-slice staging to LDS uses GLOBAL_LOAD_ASYNC_TO_LDS_B128 (ASYNCcnt
// path); the builtin on this toolchain takes typed AS-qualified int4*.
// ---------------------------------------------------------------------------

typedef __attribute__((ext_vector_type(2))) float v2f;
typedef __attribute__((ext_vector_type(8))) float v8f;
typedef __attribute__((ext_vector_type(4))) int   v4i;

// Pointer-to-AS-qualified-int4 types for the async-to-LDS builtin.
typedef __attribute__((address_space(1))) v4i* as1_v4i_ptr;
typedef __attribute__((address_space(3))) v4i* as3_v4i_ptr;

constexpr int N_VOX = 262144;
constexpr int M_PAIR = 131072;
constexpr int K3 = 27;
constexpr int CIN = 32;
constexpr int COUT = 32;

constexpr int WAVES_PER_BLOCK = 8;   // 256 threads, wave32
constexpr int CHUNKS_PER_WAVE = 8;   // 16-row tiles per wave
constexpr int ROWS_PER_BLOCK = 16 * WAVES_PER_BLOCK * CHUNKS_PER_WAVE; // 1024
constexpr int BLOCKS_PER_K = M_PAIR / ROWS_PER_BLOCK;                  // 128
static_assert(M_PAIR % ROWS_PER_BLOCK == 0, "tiling must divide M");

#ifndef __has_builtin
#define __has_builtin(x) 0
#endif

__global__ __launch_bounds__(256) void zero_f32x4(float4* __restrict__ p) {
    p[(size_t)blockIdx.x * blockDim.x + threadIdx.x] =
        make_float4(0.f, 0.f, 0.f, 0.f);
}

__global__ __launch_bounds__(256) void spconv_wmma_f32(
    const float* __restrict__ x,      // [N, CIN]
    const float* __restrict__ w,      // [K3, CIN, COUT]
    const int*   __restrict__ iidx,   // [K3, M]
    const int*   __restrict__ oidx,   // [K3, M]
    float*       __restrict__ out)    // [N, COUT]
{
    __shared__ float Bs[CIN * COUT];  // 4 KB: kernel slice for this block's k

    const int k  = blockIdx.x / BLOCKS_PER_K;
    const int kb = blockIdx.x % BLOCKS_PER_K;

    // Stage kernel[k] (32x32 fp32) into LDS: 256 threads x 16B.
#if __has_builtin(__builtin_amdgcn_global_load_async_to_lds_b128)
    {
        const float* src = w + (size_t)k * CIN * COUT + (size_t)threadIdx.x * 4;
        float*       dst = Bs + (size_t)threadIdx.x * 4;
        // Global generic pointers are numerically AS(1); generic LDS pointers
        // carry the LDS byte offset in their low 32 bits -> AS(3) value.
        __builtin_amdgcn_global_load_async_to_lds_b128(
            (as1_v4i_ptr)(uintptr_t)src,
            (as3_v4i_ptr)(uint32_t)(uintptr_t)dst,
            /*offset=*/0, /*cpol=*/0);
#if __has_builtin(__builtin_amdgcn_s_wait_asynccnt)
        __builtin_amdgcn_s_wait_asynccnt(0);
#else
        asm volatile("s_wait_asynccnt 0x0" ::: "memory");
#endif
    }
#else
    {
        const float4* src = (const float4*)(w + (size_t)k * CIN * COUT);
        ((float4*)Bs)[threadIdx.x] = src[threadIdx.x];
    }
#endif
    __syncthreads();

    const int lane   = threadIdx.x & 31;
    const int wid    = threadIdx.x >> 5;
    const int laneLo = lane & 15;   // N column within 16-wide tile / A row id
    const int laneHi = lane >> 4;   // selects K-pair (A) / M-half (C)

    // Hoist B fragments for both 16-col output tiles across all 8 K-steps.
    // ISA 4x16 f32 B layout: VGPR0 = rows K=0 (lanes 0-15) / K=2 (16-31),
    // VGPR1 = rows K=1 / K=3.
    v2f bf[2][8];
#pragma unroll
    for (int n = 0; n < 2; ++n) {
#pragma unroll
        for (int s = 0; s < 8; ++s) {
            const int kk = 4 * s + 2 * laneHi;
            bf[n][s].x = Bs[(kk    ) * COUT + 16 * n + laneLo];
            bf[n][s].y = Bs[(kk + 1) * COUT + 16 * n + laneLo];
        }
    }

    const int mWave = (kb * WAVES_PER_BLOCK + wid) * (CHUNKS_PER_WAVE * 16);
    const int base  = k * M_PAIR + mWave;

    int rowIn = iidx[base + laneLo];   // gather row for chunk 0

#pragma unroll 1
    for (int c = 0; c < CHUNKS_PER_WAVE; ++c) {
        const float* xr = x + (size_t)rowIn * CIN;

        // Prefetch next chunk's gather row (emits global_prefetch_b8).
        if (c + 1 < CHUNKS_PER_WAVE) {
            const int rowNext = iidx[base + (c + 1) * 16 + laneLo];
            __builtin_prefetch(x + (size_t)rowNext * CIN, 0, 0);
            rowIn = rowNext;
        }

        v8f acc0 = {};
        v8f acc1 = {};
        // ISA 16x4 f32 A layout: lane L holds row M=L%16, elems
        // {4s + 2*(L/16), +1} -> one b64 load per lane per step.
#pragma unroll
        for (int s = 0; s < 8; ++s) {
            const v2f a = *(const v2f*)(xr + 4 * s + 2 * laneHi);
            acc0 = __builtin_amdgcn_wmma_f32_16x16x4_f32(
                false, a, false, bf[0][s], (short)0, acc0, false, false);
            acc1 = __builtin_amdgcn_wmma_f32_16x16x4_f32(
                false, a, false, bf[1][s], (short)0, acc1, false, false);
        }

        // C/D layout: VGPR v, lane L -> (M = v + 8*(L/16), N = L%16).
        const int ob = base + c * 16 + 8 * laneHi;
#pragma unroll
        for (int v = 0; v < 8; ++v) {
            const int orow = oidx[ob + v];
            float* op = out + (size_t)orow * COUT;
            atomicAdd(op + laneLo,      acc0[v]);
            atomicAdd(op + 16 + laneLo, acc1[v]);
        }
    }
}

extern "C" void kernel_launch(void* const* d_in, const int* in_sizes, int n_in,
                              void* d_out, int out_size, void* d_ws, size_t ws_size,
                              hipStream_t stream) {
    const float* x    = (const float*)d_in[0];
    const float* w    = (const float*)d_in[1];
    const int*   iidx = (const int*)d_in[2];
    const int*   oidx = (const int*)d_in[3];
    float*       out  = (float*)d_out;

    // Zero the scatter-add destination (harness poisons d_out).
    {
        const int n4 = out_size / 4;              // out_size = N*COUT, /4 per float4
        const int blocks = (n4 + 255) / 256;
        zero_f32x4<<<blocks, 256, 0, stream>>>((float4*)out);
    }

    const int grid = K3 * BLOCKS_PER_K;           // 27 * 128 = 3456
    spconv_wmma_f32<<<grid, 256, 0, stream>>>(x, w, iidx, oidx, out);
}